// ROIPool_42966852829222
// MI455X (gfx1250) — compile-verified
//
#include <hip/hip_runtime.h>
#include <cfloat>

// ROI max-pool, reference shapes: x (N=2, C=256, H=64, W=64) f32, rois (K=128, 5) f32,
// out (K, C, 7, 7) f32. SPATIAL_SCALE = 0.0625, pooled 7x7.
//
// Design (MI455X, gfx1250, wave32):
//  * One thread per output element -> 1.6M threads (50K wave32 waves): latency-bound
//    gather problem, input (8 MB) is L2-resident (192 MB L2), so no LDS tiling of
//    pixel data is warranted (bins partition the rect; reuse ~1x).
//  * Per 256-thread block the ROI index k is uniform (C*7*7 = 12544 = 49*256), so the
//    5-float ROI descriptor is staged once per block into LDS via the CDNA5 async
//    global->LDS path (ASYNCcnt) and broadcast to all 8 waves.
//  * Output is streamed with a non-temporal store (TH=NT) to keep L2 for the input.

#define POOLED   7
#define CHANNELS 256
#define HEIGHT   64
#define WIDTH    64
#define SCALE    0.0625f

__global__ __launch_bounds__(256) void ROIPool_42966852829222_kernel(
    const float* __restrict__ x,
    const float* __restrict__ rois,
    float* __restrict__ out)
{
    __shared__ float s_roi[8];   // 5 used; padded

    const int t   = threadIdx.x;
    const int idx = blockIdx.x * 256 + t;                 // < K*C*49, exact multiple of 256
    const int per_roi = CHANNELS * POOLED * POOLED;       // 12544 = 49 * 256
    const int k = idx / per_roi;                          // uniform across the block

    // ---- CDNA5 async global->LDS broadcast of this block's ROI descriptor ----
    if (t < 5) {
        unsigned long long gaddr =
            (unsigned long long)(const void*)(rois + (size_t)k * 5 + t);
        // Low 32 bits of a flat pointer into LDS == wave-relative LDS byte offset.
        unsigned int laddr =
            (unsigned int)(unsigned long long)(void*)&s_roi[t];
        asm volatile("global_load_async_to_lds_b32 %0, %1, off"
                     :: "v"(laddr), "v"(gaddr)
                     : "memory");
    }
#if __has_builtin(__builtin_amdgcn_s_wait_asynccnt)
    __builtin_amdgcn_s_wait_asynccnt(0);
#else
    asm volatile("s_wait_asynccnt 0" ::: "memory");
#endif
    __syncthreads();

    // ---- decode output coordinate ----
    const int rem = idx % per_roi;
    const int c   = rem / (POOLED * POOLED);
    const int b   = rem % (POOLED * POOLED);
    const int ph  = b / POOLED;
    const int pw  = b % POOLED;

    // ---- ROI geometry (matches jnp.round = round-half-even via rintf) ----
    const int n  = (int)s_roi[0];
    const int x1 = (int)rintf(s_roi[1] * SCALE);
    const int y1 = (int)rintf(s_roi[2] * SCALE);
    const int x2 = (int)rintf(s_roi[3] * SCALE);
    const int y2 = (int)rintf(s_roi[4] * SCALE);

    const int roi_w = max(x2 - x1 + 1, 1);
    const int roi_h = max(y2 - y1 + 1, 1);
    const float bw = (float)roi_w / 7.0f;   // f32 division, same as reference
    const float bh = (float)roi_h / 7.0f;

    int ws = (int)floorf((float)pw * bw) + x1;
    int we = (int)ceilf((float)(pw + 1) * bw) + x1;
    int hs = (int)floorf((float)ph * bh) + y1;
    int he = (int)ceilf((float)(ph + 1) * bh) + y1;
    ws = min(max(ws, 0), WIDTH);
    we = min(max(we, 0), WIDTH);
    hs = min(max(hs, 0), HEIGHT);
    he = min(max(he, 0), HEIGHT);
    const bool empty = (he <= hs) || (we <= ws);

    // ---- max-reduce the bin ----
    const float* __restrict__ plane =
        x + ((size_t)n * CHANNELS + c) * (size_t)(HEIGHT * WIDTH);
    float m = -FLT_MAX;   // == jnp.finfo(float32).min
    for (int h = hs; h < he; ++h) {
        const float* __restrict__ row = plane + h * WIDTH;
        for (int w = ws; w < we; ++w) {
            m = fmaxf(m, row[w]);
        }
    }

    // Streamed output: non-temporal store keeps L2 for the (reused) input planes.
    __builtin_nontemporal_store(empty ? 0.0f : m, &out[idx]);
}

extern "C" void kernel_launch(void* const* d_in, const int* in_sizes, int n_in,
                              void* d_out, int out_size, void* d_ws, size_t ws_size,
                              hipStream_t stream) {
    const float* x    = (const float*)d_in[0];
    const float* rois = (const float*)d_in[1];
    float* out        = (float*)d_out;

    const int K     = in_sizes[1] / 5;                       // 128
    const int total = K * CHANNELS * POOLED * POOLED;        // == out_size, multiple of 256
    const int blocks = total / 256;                          // exact (12544 = 49*256)

    ROIPool_42966852829222_kernel<<<blocks, 256, 0, stream>>>(x, rois, out);
}